// Backbone_67937792688557
// MI455X (gfx1250) — compile-verified
//
#include <hip/hip_runtime.h>

#define N_NODES 50000
#define N_EDGES 600000
#define HID 128
#define NEG_SLOPE 0.01f
#define N_LAYERS 3

typedef __attribute__((ext_vector_type(2))) float v2f;
typedef __attribute__((ext_vector_type(8))) float v8f;

// ---------------------------------------------------------------------------
// Zero the aggregation buffer (float4 stores, fully coalesced)
// ---------------------------------------------------------------------------
__global__ __launch_bounds__(256) void zero_kernel(float4* __restrict__ p, int n4) {
  int i = blockIdx.x * 256 + threadIdx.x;
  if (i < n4) p[i] = make_float4(0.f, 0.f, 0.f, 0.f);
}

// ---------------------------------------------------------------------------
// Edge phase: msg = relu(h[src] + edge_attr @ Wl + bl); agg[dst] += msg
// One wave (32 lanes) per edge, 4 channels per lane (float4).
// h[src] gather is 16B/lane coalesced; scatter uses HW f32 atomics.
// ---------------------------------------------------------------------------
__global__ __launch_bounds__(256) void edge_kernel(
    const float* __restrict__ h,
    const int*   __restrict__ src,
    const int*   __restrict__ dst,
    const float* __restrict__ ea,    // [E,2]
    const float* __restrict__ Wl0,   // [128] (row 0 of this layer's edge proj)
    const float* __restrict__ Wl1,   // [128] (row 1)
    const float* __restrict__ bl,    // [128]
    float* __restrict__ agg)         // [N,128]
{
  int tid = blockIdx.x * 256 + threadIdx.x;
  int e   = tid >> 5;          // edge id (wave-uniform)
  int c   = (tid & 31) * 4;    // channel base for this lane
  if (e >= N_EDGES) return;

  int   s  = src[e];
  int   d  = dst[e];
  float a0 = ea[2 * e];
  float a1 = ea[2 * e + 1];

  float4 w0 = *(const float4*)(Wl0 + c);   // layer constants: L0/L2 resident
  float4 w1 = *(const float4*)(Wl1 + c);
  float4 bb = *(const float4*)(bl  + c);
  float4 hv = *(const float4*)(h + (long)s * HID + c);

  float m0 = hv.x + a0 * w0.x + a1 * w1.x + bb.x;
  float m1 = hv.y + a0 * w0.y + a1 * w1.y + bb.y;
  float m2 = hv.z + a0 * w0.z + a1 * w1.z + bb.z;
  float m3 = hv.w + a0 * w0.w + a1 * w1.w + bb.w;
  m0 = m0 > 0.f ? m0 : 0.f;
  m1 = m1 > 0.f ? m1 : 0.f;
  m2 = m2 > 0.f ? m2 : 0.f;
  m3 = m3 > 0.f ? m3 : 0.f;

  float* ap = agg + (long)d * HID + c;
  unsafeAtomicAdd(ap + 0, m0);   // -> global_atomic_add_f32 (no CAS loop)
  unsafeAtomicAdd(ap + 1, m1);
  unsafeAtomicAdd(ap + 2, m2);
  unsafeAtomicAdd(ap + 3, m3);
}

// ---------------------------------------------------------------------------
// Node phase: out = leakyrelu((h + agg) @ W + b)
// M=50000, K=128, N=128.  One wave per 16x16 output tile using full-precision
// V_WMMA_F32_16X16X4_F32 (K-loop of 32 accumulating WMMAs).
// Block = 8 waves covering the full N=128 for one 16-row m-tile.
//
// VGPR layouts (ISA 7.12.2, 32-bit):
//   A 16x4 : lane m=lane&15, half=lane>>4 ; VGPR j = A[m][half*2 + j]
//   B 4x16 : VGPR v ; lanes 0-15 row K=v, lanes 16-31 row K=v+2
//   C 16x16: VGPR v ; lanes 0-15 M=v,  lanes 16-31 M=v+8, N=lane&15
// ---------------------------------------------------------------------------
__global__ __launch_bounds__(256) void gine_gemm_kernel(
    const float* __restrict__ h,     // [N,128]
    const float* __restrict__ agg,   // [N,128]
    const float* __restrict__ W,     // [128,128] this layer
    const float* __restrict__ bias,  // [128]     this layer
    float* __restrict__ out)         // [N,128]
{
  int wave = threadIdx.x >> 5;      // 0..7 -> n tile
  int lane = threadIdx.x & 31;
  int m0   = blockIdx.x * 16;
  int n0   = wave * 16;
  int m    = lane & 15;
  int half = lane >> 4;
  int n    = n0 + m;                // B/D column for this lane

  const float* xr = h   + (long)(m0 + m) * HID + half * 2;
  const float* gr = agg + (long)(m0 + m) * HID + half * 2;

  v8f c = {};
  for (int kk = 0; kk < HID; kk += 4) {
    float2 hx = *(const float2*)(xr + kk);
    float2 gx = *(const float2*)(gr + kk);
    v2f a;
    a.x = hx.x + gx.x;              // fused X = h + agg (eps = 0)
    a.y = hx.y + gx.y;

    int k0 = kk + half * 2;
    v2f bm;
    bm.x = W[(long)k0 * HID + n];         // row K = k0   (VGPR0)
    bm.y = W[(long)(k0 + 1) * HID + n];   // row K = k0+1 (VGPR1)

    // D = A x B + C, fp32 accumulate chain
    c = __builtin_amdgcn_wmma_f32_16x16x4_f32(
        /*neg_a=*/false, a, /*neg_b=*/false, bm,
        /*c_mod=*/(short)0, c, /*reuse_a=*/false, /*reuse_b=*/false);
  }

  float bv = bias[n];
#pragma unroll
  for (int v = 0; v < 8; ++v) {
    int row = m0 + v + half * 8;
    float y = c[v] + bv;
    out[(long)row * HID + n] = (y >= 0.f) ? y : NEG_SLOPE * y;
  }
}

// ---------------------------------------------------------------------------
extern "C" void kernel_launch(void* const* d_in, const int* in_sizes, int n_in,
                              void* d_out, int out_size, void* d_ws, size_t ws_size,
                              hipStream_t stream) {
  (void)in_sizes; (void)n_in; (void)out_size; (void)ws_size;

  const float* x  = (const float*)d_in[0];          // [N,128]
  const int*   ei = (const int*)  d_in[1];          // [2,E] (int32 on device)
  const float* ea = (const float*)d_in[2];          // [E,2]
  const float* Wl = (const float*)d_in[3];          // [3,2,128]
  const float* bl = (const float*)d_in[4];          // [3,128]
  const float* W  = (const float*)d_in[5];          // [3,128,128]
  const float* b  = (const float*)d_in[6];          // [3,128]
  float*       out = (float*)d_out;

  const int* src = ei;
  const int* dst = ei + N_EDGES;

  const long NH = (long)N_NODES * HID;              // 6.4M floats
  float* agg = (float*)d_ws;                        // 25.6 MB
  float* h0  = agg + NH;                            // 25.6 MB
  float* h1  = h0 + NH;                             // 25.6 MB

  const float* hin = x;
  for (int layer = 0; layer < N_LAYERS; ++layer) {
    float* hout = (layer == N_LAYERS - 1) ? out : (layer == 0 ? h0 : h1);

    zero_kernel<<<(int)(NH / 4 + 255) / 256, 256, 0, stream>>>((float4*)agg, (int)(NH / 4));

    edge_kernel<<<N_EDGES / 8, 256, 0, stream>>>(
        hin, src, dst, ea,
        Wl + (long)layer * 2 * HID,          // Wl[layer][0][:]
        Wl + (long)layer * 2 * HID + HID,    // Wl[layer][1][:]
        bl + (long)layer * HID,
        agg);

    gine_gemm_kernel<<<N_NODES / 16, 256, 0, stream>>>(
        hin, agg,
        W + (long)layer * HID * HID,
        b + (long)layer * HID,
        hout);

    hin = hout;
  }
}